// GraphSAGE_pia_26998164422767
// MI455X (gfx1250) — compile-verified
//
#include <hip/hip_runtime.h>
#include <hip/hip_bf16.h>

// GraphSAGE 3-layer forward for MI455X (gfx1250, wave32, WMMA).
//
// d_in: 0=feat[N,64] f32, 1=edge_index[2,E] int32, then per layer l:
//       W_self{l}[din,dout], W_neigh{l}[din,dout], b{l}[dout]  (all f32)
// d_out: h_final[N,16] ++ embed0[N,64] ++ embed1[N,64]  (f32)
// d_ws:  agg[N*64] f32, deg/invdeg[N] f32

typedef __attribute__((ext_vector_type(2))) float v2f;
typedef __attribute__((ext_vector_type(8))) float v8f;

#define NODE_FEATS 64  // din is 64 for every layer

// ---------------------------------------------------------------- utilities
__global__ void zero_f32(float* __restrict__ p, long long n) {
    long long i = (long long)blockIdx.x * blockDim.x + threadIdx.x;
    long long stride = (long long)gridDim.x * blockDim.x;
    for (; i < n; i += stride) p[i] = 0.0f;
}

__global__ void count_deg(const int* __restrict__ dst, float* __restrict__ deg, int n_edges) {
    int e = blockIdx.x * blockDim.x + threadIdx.x;
    if (e < n_edges) atomicAdd(&deg[dst[e]], 1.0f);
}

__global__ void make_invdeg(float* __restrict__ deg, int n) {
    int i = blockIdx.x * blockDim.x + threadIdx.x;
    if (i < n) deg[i] = 1.0f / fmaxf(deg[i], 1.0f);
}

// -------------------------------------------------- edge scatter (agg += h[src])
// 16 threads per edge, each moves a float4 (64 feats). relu_in applies the
// previous layer's activation at read time (raw pre-ReLU tensors live in d_out).
__global__ void sage_scatter(const float* __restrict__ h,
                             const int* __restrict__ src, const int* __restrict__ dst,
                             float* __restrict__ agg, int n_edges, int relu_in) {
    int idx = blockIdx.x * blockDim.x + threadIdx.x;
    int total = n_edges * 16;
    if (idx >= total) return;
    int e = idx >> 4;
    int c = idx & 15;
    int s = src[e];
    int d = dst[e];
    float4 v = ((const float4*)(h + (long long)s * NODE_FEATS))[c];
    if (relu_in) {
        v.x = fmaxf(v.x, 0.0f); v.y = fmaxf(v.y, 0.0f);
        v.z = fmaxf(v.z, 0.0f); v.w = fmaxf(v.w, 0.0f);
    }
    float* ap = agg + (long long)d * NODE_FEATS + c * 4;
    atomicAdd(ap + 0, v.x);
    atomicAdd(ap + 1, v.y);
    atomicAdd(ap + 2, v.z);
    atomicAdd(ap + 3, v.w);
}

// -------------------------------------------------- fused dual-GEMM epilogue
// O[16x16 tile] = relu?(H) @ Ws + (AGG * invdeg_row) @ Wn + b
// One wave per 16x16 output tile; V_WMMA_F32_16X16X4_F32, K=64 in 16 steps,
// two WMMAs per step (self + neigh) chained through one accumulator.
__global__ void sage_gemm(const float* __restrict__ H, const float* __restrict__ AGG,
                          const float* __restrict__ invdeg,
                          const float* __restrict__ Ws, const float* __restrict__ Wn,
                          const float* __restrict__ bias,
                          float* __restrict__ O,
                          int row_tiles, int col_tiles, int dout, int relu_in) {
    const int wave = blockIdx.x * (blockDim.x >> 5) + (threadIdx.x >> 5);  // wave-uniform
    if (wave >= row_tiles * col_tiles) return;                             // uniform exit
    const int lane = threadIdx.x & 31;
    const int rt = wave / col_tiles;
    const int ct = wave % col_tiles;
    const int row0 = rt << 4;
    const int col0 = ct << 4;

    const int m = lane & 15;          // A: row within tile / B: col within tile
    const int khalf = (lane >> 4) * 2;  // lane-half K offset

    const float* __restrict__ hrow = H   + (long long)(row0 + m) * NODE_FEATS;
    const float* __restrict__ arow = AGG + (long long)(row0 + m) * NODE_FEATS;
    const float idg = invdeg[row0 + m];

    v8f acc = {};
#pragma unroll
    for (int k0 = 0; k0 < NODE_FEATS; k0 += 4) {
        const int k = k0 + khalf;
        v2f a, an, bS, bN;
        float a0 = hrow[k], a1 = hrow[k + 1];
        if (relu_in) { a0 = fmaxf(a0, 0.0f); a1 = fmaxf(a1, 0.0f); }
        a.x = a0; a.y = a1;
        an.x = arow[k] * idg; an.y = arow[k + 1] * idg;
        // W is [64 x dout] row-major: B fragment = rows k,k+1 of W, col col0+m
        bS.x = Ws[(long long)k * dout + col0 + m];
        bS.y = Ws[(long long)(k + 1) * dout + col0 + m];
        bN.x = Wn[(long long)k * dout + col0 + m];
        bN.y = Wn[(long long)(k + 1) * dout + col0 + m];
        acc = __builtin_amdgcn_wmma_f32_16x16x4_f32(false, a,  false, bS, (short)0, acc, false, false);
        acc = __builtin_amdgcn_wmma_f32_16x16x4_f32(false, an, false, bN, (short)0, acc, false, false);
    }

    // C/D layout: component v -> row row0 + v + 8*(lane>>4), col col0 + (lane&15)
    const float bval = bias[col0 + m];
    const int rbase = row0 + ((lane >> 4) << 3);
#pragma unroll
    for (int v = 0; v < 8; ++v) {
        O[(long long)(rbase + v) * dout + col0 + m] = acc[v] + bval;
    }
}

// ---------------------------------------------------------------- launcher
extern "C" void kernel_launch(void* const* d_in, const int* in_sizes, int n_in,
                              void* d_out, int out_size, void* d_ws, size_t ws_size,
                              hipStream_t stream) {
    const float* feat = (const float*)d_in[0];
    const int*   ei   = (const int*)d_in[1];
    const float* Ws0 = (const float*)d_in[2];
    const float* Wn0 = (const float*)d_in[3];
    const float* b0  = (const float*)d_in[4];
    const float* Ws1 = (const float*)d_in[5];
    const float* Wn1 = (const float*)d_in[6];
    const float* b1  = (const float*)d_in[7];
    const float* Ws2 = (const float*)d_in[8];
    const float* Wn2 = (const float*)d_in[9];
    const float* b2  = (const float*)d_in[10];

    const int N = in_sizes[0] / NODE_FEATS;    // 100000
    const int E = in_sizes[1] / 2;             // 1600000
    const int* src = ei;
    const int* dst = ei + E;

    const int N_CLASSES = 16;
    float* outF = (float*)d_out;                       // [N,16] final
    float* out0 = outF + (long long)N * N_CLASSES;     // [N,64] embed0 (raw layer-0)
    float* out1 = out0 + (long long)N * NODE_FEATS;    // [N,64] embed1 (raw layer-1)

    float* agg = (float*)d_ws;                         // [N*64]
    float* deg = agg + (long long)N * NODE_FEATS;      // [N] -> becomes invdeg

    const int row_tiles = N / 16;  // 6250 (N divisible by 16)

    // degree (layer-invariant): zero agg+deg together, count, invert
    {
        long long nz = (long long)N * NODE_FEATS + N;
        zero_f32<<<2048, 256, 0, stream>>>(agg, nz);  // grid-stride; also zeroes deg
        count_deg<<<(E + 255) / 256, 256, 0, stream>>>(dst, deg, E);
        make_invdeg<<<(N + 255) / 256, 256, 0, stream>>>(deg, N);
    }

    const int scat_total = E * 16;
    const int scat_blocks = (scat_total + 255) / 256;

    // ---- layer 0: input = feat (no relu), dout=64 -> out0 (embed0, raw)
    sage_scatter<<<scat_blocks, 256, 0, stream>>>(feat, src, dst, agg, E, 0);
    {
        int col_tiles = NODE_FEATS / 16;
        int waves = row_tiles * col_tiles;
        sage_gemm<<<(waves + 7) / 8, 256, 0, stream>>>(feat, agg, deg, Ws0, Wn0, b0,
                                                       out0, row_tiles, col_tiles, NODE_FEATS, 0);
    }

    // ---- layer 1: input = relu(out0), dout=64 -> out1 (embed1, raw)
    zero_f32<<<2048, 256, 0, stream>>>(agg, (long long)N * NODE_FEATS);
    sage_scatter<<<scat_blocks, 256, 0, stream>>>(out0, src, dst, agg, E, 1);
    {
        int col_tiles = NODE_FEATS / 16;
        int waves = row_tiles * col_tiles;
        sage_gemm<<<(waves + 7) / 8, 256, 0, stream>>>(out0, agg, deg, Ws1, Wn1, b1,
                                                       out1, row_tiles, col_tiles, NODE_FEATS, 1);
    }

    // ---- layer 2: input = relu(out1), dout=16 -> outF (logits)
    zero_f32<<<2048, 256, 0, stream>>>(agg, (long long)N * NODE_FEATS);
    sage_scatter<<<scat_blocks, 256, 0, stream>>>(out1, src, dst, agg, E, 1);
    {
        int col_tiles = N_CLASSES / 16;  // 1
        int waves = row_tiles * col_tiles;
        sage_gemm<<<(waves + 7) / 8, 256, 0, stream>>>(out1, agg, deg, Ws2, Wn2, b2,
                                                       outF, row_tiles, col_tiles, N_CLASSES, 1);
    }
}